// FixedDeleteGate_83502754169446
// MI455X (gfx1250) — compile-verified
//
#include <hip/hip_runtime.h>
#include <stdint.h>

#define S_LEN 8192
#define B_DIM 8
#define NT    1024
#define NCHNK (S_LEN / NT)   // 8

// 256-bit membership bitmap for SEP tokens:
// {1,12,13,35..50,61,62,63} -> word0; {64..67,94..99,126,127} -> word1; {128,129} -> word2
__device__ __forceinline__ unsigned is_sep_tok(int v) {
  if ((unsigned)v >= 256u) return 0u;
  unsigned long long w;
  switch (v >> 6) {
    case 0:  w = 0xE007FFF800003002ull; break;
    case 1:  w = 0xC000000FC000000Full; break;
    case 2:  w = 0x0000000000000003ull; break;
    default: w = 0ull; break;
  }
  return (unsigned)((w >> (v & 63)) & 1ull);
}

// wave32 inclusive scan via shuffles
__device__ __forceinline__ unsigned wave_incl_scan(unsigned x, int lane) {
#pragma unroll
  for (int d = 1; d < 32; d <<= 1) {
    unsigned t = (unsigned)__shfl_up((int)x, d, 32);
    if (lane >= d) x += t;
  }
  return x;
}

__global__ __launch_bounds__(NT, 1)
void FixedDeleteGate_kernel(const int* __restrict__ ids, float* __restrict__ out) {
  __shared__ unsigned s_cnt[S_LEN + 1];   // bincount bins -> inclusive cumsum (in place)
  __shared__ int      s_ids[S_LEN];       // async-staged input_ids row (32 KB)
  __shared__ unsigned s_wave[33];         // per-wave partial sums + chunk total

  const int tid  = threadIdx.x;
  const int lane = tid & 31;
  const int wv   = tid >> 5;

  // ---- zero bins ----
  for (int i = tid; i <= S_LEN; i += NT) s_cnt[i] = 0u;
  __syncthreads();

  const unsigned long long base = (unsigned long long)(uintptr_t)ids;

  // ---- Phase B: per-row inclusive scan of is_sep + global bincount (LDS atomics) ----
  for (int b = 0; b < B_DIM; ++b) {
    // Stage the whole row into LDS with CDNA5 async-to-LDS loads (8 in flight per lane,
    // single ASYNCcnt wait -> one memory round-trip per row instead of eight).
#pragma unroll
    for (int c = 0; c < NCHNK; ++c) {
      const int idx = c * NT + tid;
      const unsigned lds_dst = (unsigned)(uintptr_t)&s_ids[idx];           // low 32 bits = LDS offset
      const unsigned voff    = (unsigned)((b * S_LEN + idx) * 4);
      asm volatile("global_load_async_to_lds_b32 %0, %1, %2"
                   :: "v"(lds_dst), "v"(voff), "s"(base)
                   : "memory");
    }
    // Prefetch next row into cache while this row's async loads are in flight.
    if (b + 1 < B_DIM && tid < 512)
      __builtin_prefetch(ids + (b + 1) * S_LEN + tid * 16, 0, 0);  // 512 * 64B = full 32KB row
    asm volatile("s_wait_asynccnt 0" ::: "memory");
    __syncthreads();

    unsigned carry = 0u;
    for (int c = 0; c < NCHNK; ++c) {
      const unsigned x   = is_sep_tok(s_ids[c * NT + tid]);
      const unsigned inc = wave_incl_scan(x, lane);
      if (lane == 31) s_wave[wv] = inc;
      __syncthreads();
      if (wv == 0) {
        unsigned w = wave_incl_scan(s_wave[lane], lane);
        s_wave[lane] = w;                    // inclusive per-wave sums
        if (lane == 31) s_wave[32] = w;      // chunk total
      }
      __syncthreads();
      const unsigned sp = inc + (wv ? s_wave[wv - 1] : 0u) + carry;  // inclusive cumsum
      atomicAdd(&s_cnt[sp], 1u);
      carry += s_wave[32];
      __syncthreads();
    }
  }
  __syncthreads();

  // ---- Phase C: inclusive scan of counts[0..S] -> cum (in place; per-thread read/write same slot) ----
  unsigned carry2 = 0u;
  for (int c = 0; c < NCHNK + 1; ++c) {     // 9 chunks cover 8193 bins
    const int  idx = c * NT + tid;
    const bool act = (idx <= S_LEN);
    const unsigned x   = act ? s_cnt[idx] : 0u;
    const unsigned inc = wave_incl_scan(x, lane);
    if (lane == 31) s_wave[wv] = inc;
    __syncthreads();
    if (wv == 0) {
      unsigned w = wave_incl_scan(s_wave[lane], lane);
      s_wave[lane] = w;
      if (lane == 31) s_wave[32] = w;
    }
    __syncthreads();
    const unsigned val = inc + (wv ? s_wave[wv - 1] : 0u) + carry2;
    if (act) s_cnt[idx] = val;
    carry2 += s_wave[32];
    __syncthreads();
  }

  // ---- Phase D: per-position segment lookup + broadcast writes ----
  for (int p = tid; p < S_LEN; p += NT) {
    // first j with cum[j] > p  (== reference's searchsorted(starts, p, 'right') - 1)
    int lo = 0, hi = S_LEN + 1;
    while (lo < hi) {
      const int mid = (lo + hi) >> 1;
      if (s_cnt[mid] > (unsigned)p) hi = mid; else lo = mid + 1;
    }
    const int      j   = lo;                              // cum[S_LEN] = B*S > p, so j <= S_LEN
    const unsigned st  = j ? s_cnt[j - 1] : 0u;
    const unsigned cnt = s_cnt[j] - st;                   // > 0 by construction
    const unsigned nz  = (cnt + 1u) >> 1;                 // ceil(0.5 * count)
    const float v = (((unsigned)p - st) < nz) ? 0.0f : -30.0f;
#pragma unroll
    for (int b = 0; b < B_DIM; ++b) {
      out[b * S_LEN + p]                     = v;         // output 0: mask
      out[B_DIM * S_LEN + b * S_LEN + p]     = v;         // output 1: same mask
    }
  }
}

extern "C" void kernel_launch(void* const* d_in, const int* in_sizes, int n_in,
                              void* d_out, int out_size, void* d_ws, size_t ws_size,
                              hipStream_t stream) {
  (void)in_sizes; (void)n_in; (void)out_size; (void)d_ws; (void)ws_size;
  const int* input_ids = (const int*)d_in[1];   // d_in[0] = hidden_states (unused by reference output)
  float* out = (float*)d_out;                   // 2 * 8 * 8192 floats
  FixedDeleteGate_kernel<<<1, NT, 0, stream>>>(input_ids, out);
}